// UpsampleLoss_17867063951814
// MI455X (gfx1250) — compile-verified
//
#include <hip/hip_runtime.h>
#include <math.h>

typedef __attribute__((ext_vector_type(2))) float v2f;
typedef __attribute__((ext_vector_type(8))) float v8f;

#define ALPHA_C 1.0f
#define RADIUS_C 0.07f
#define H_C 0.03f
#define EPS_C 1e-12f
#define BIGF 3.402823e38f

// min across each 16-lane half of the wave32 (butterfly stays inside the half)
__device__ __forceinline__ float half_min(float x) {
    x = fminf(x, __shfl_xor(x, 1, 32));
    x = fminf(x, __shfl_xor(x, 2, 32));
    x = fminf(x, __shfl_xor(x, 4, 32));
    x = fminf(x, __shfl_xor(x, 8, 32));
    return x;
}

// ---------------------------------------------------------------------------
// Chamfer one direction: for each 16-row tile of A, min over all of B of
// ||a-b||^2, computed as a2 + b2 - 2*dot via V_WMMA_F32_16X16X4_F32 (K=3
// padded to 4 with a zero column). One wave per tile. Writes per-tile partial
// sums (sum of 16 row-mins) to partials[blockIdx.x].
// ---------------------------------------------------------------------------
__global__ void __launch_bounds__(32)
chamfer_rowmin_kernel(const float* __restrict__ A, const float* __restrict__ Bm,
                      float* __restrict__ partials, int N) {
    const int lane = threadIdx.x;        // 0..31
    const int half = lane >> 4;          // 0 or 1
    const int l    = lane & 15;
    const int ntiles = N >> 4;
    const int b     = blockIdx.x / ntiles;
    const int rtile = blockIdx.x % ntiles;
    const int row0  = rtile << 4;

    const float* Ab = A  + ((size_t)b * N + row0) * 3;
    const float* Bb = Bm + (size_t)b * N * 3;

    // A fragment: lanes 0-15 -> (x,y) of row l; lanes 16-31 -> (z,0) of row l
    v2f afrag;
    afrag[0] = (half == 0) ? Ab[l * 3 + 0] : Ab[l * 3 + 2];
    afrag[1] = (half == 0) ? Ab[l * 3 + 1] : 0.0f;

    // squared norms of this lane's 8 output rows (m = v + 8*half)
    float a2[8];
#pragma unroll
    for (int v = 0; v < 8; ++v) {
        const float* p = Ab + (v + 8 * half) * 3;
        a2[v] = p[0] * p[0] + p[1] * p[1] + p[2] * p[2];
    }

    float rmin[8];
#pragma unroll
    for (int v = 0; v < 8; ++v) rmin[v] = BIGF;

    for (int ct = 0; ct < ntiles; ++ct) {
        const float* Bt = Bb + (size_t)(ct << 4) * 3;
        __builtin_prefetch(Bt + 48, 0, 3);   // next 16-point tile -> global_prefetch_b8

        v2f bfrag;
        bfrag[0] = (half == 0) ? Bt[l * 3 + 0] : Bt[l * 3 + 2];
        bfrag[1] = (half == 0) ? Bt[l * 3 + 1] : 0.0f;

        const float bx = Bt[l * 3 + 0], by = Bt[l * 3 + 1], bz = Bt[l * 3 + 2];
        const float b2 = bx * bx + by * by + bz * bz;   // column n = l

        v8f acc = {};
        acc = __builtin_amdgcn_wmma_f32_16x16x4_f32(
            false, afrag, false, bfrag, (short)0, acc, false, false);

#pragma unroll
        for (int v = 0; v < 8; ++v) {
            float d2 = fmaxf(a2[v] + b2 - 2.0f * acc[v], 0.0f);
            rmin[v] = fminf(rmin[v], d2);
        }
    }

    float s = 0.0f;
#pragma unroll
    for (int v = 0; v < 8; ++v) s += half_min(rmin[v]);   // half0: rows 0-7, half1: rows 8-15
    s += __shfl_xor(s, 16, 32);                           // combine halves
    if (lane == 0) partials[blockIdx.x] = s;
}

// ---------------------------------------------------------------------------
// Repulsion: pred vs pred; per row keep the 4 smallest off-diagonal d2,
// then sum (RADIUS - dist) * exp(-d2/H^2). Same WMMA tiling; per-lane top-4
// candidate lists are merged through LDS.
// ---------------------------------------------------------------------------
__global__ void __launch_bounds__(32)
repulsion_kernel(const float* __restrict__ P, float* __restrict__ partials, int N) {
    const int lane = threadIdx.x;
    const int half = lane >> 4;
    const int l    = lane & 15;
    const int ntiles = N >> 4;
    const int b     = blockIdx.x / ntiles;
    const int rtile = blockIdx.x % ntiles;
    const int row0  = rtile << 4;

    const float* Ab = P + ((size_t)b * N + row0) * 3;
    const float* Bb = P + (size_t)b * N * 3;

    v2f afrag;
    afrag[0] = (half == 0) ? Ab[l * 3 + 0] : Ab[l * 3 + 2];
    afrag[1] = (half == 0) ? Ab[l * 3 + 1] : 0.0f;

    float a2[8];
#pragma unroll
    for (int v = 0; v < 8; ++v) {
        const float* p = Ab + (v + 8 * half) * 3;
        a2[v] = p[0] * p[0] + p[1] * p[1] + p[2] * p[2];
    }

    float nn[8][4];                 // sorted ascending per row
#pragma unroll
    for (int v = 0; v < 8; ++v)
#pragma unroll
        for (int j = 0; j < 4; ++j) nn[v][j] = BIGF;

    for (int ct = 0; ct < ntiles; ++ct) {
        const float* Bt = Bb + (size_t)(ct << 4) * 3;
        __builtin_prefetch(Bt + 48, 0, 3);

        v2f bfrag;
        bfrag[0] = (half == 0) ? Bt[l * 3 + 0] : Bt[l * 3 + 2];
        bfrag[1] = (half == 0) ? Bt[l * 3 + 1] : 0.0f;

        const float bx = Bt[l * 3 + 0], by = Bt[l * 3 + 1], bz = Bt[l * 3 + 2];
        const float b2 = bx * bx + by * by + bz * bz;
        const int gcol = (ct << 4) + l;

        v8f acc = {};
        acc = __builtin_amdgcn_wmma_f32_16x16x4_f32(
            false, afrag, false, bfrag, (short)0, acc, false, false);

#pragma unroll
        for (int v = 0; v < 8; ++v) {
            const int grow = row0 + v + 8 * half;
            float d2 = fmaxf(a2[v] + b2 - 2.0f * acc[v], 0.0f);
            if (grow != gcol && d2 < nn[v][3]) {          // insertion into sorted 4-list
                nn[v][3] = d2;
                if (nn[v][3] < nn[v][2]) { float t = nn[v][2]; nn[v][2] = nn[v][3]; nn[v][3] = t; }
                if (nn[v][2] < nn[v][1]) { float t = nn[v][1]; nn[v][1] = nn[v][2]; nn[v][2] = t; }
                if (nn[v][1] < nn[v][0]) { float t = nn[v][0]; nn[v][0] = nn[v][1]; nn[v][1] = t; }
            }
        }
    }

    // merge per-lane candidates: true row top-4 is contained in the union of
    // each lane's top-4 (each lane covered columns n ≡ l (mod 16))
    __shared__ float cand[16][64];
#pragma unroll
    for (int v = 0; v < 8; ++v) {
        const int row = v + 8 * half;
#pragma unroll
        for (int j = 0; j < 4; ++j) cand[row][l * 4 + j] = nn[v][j];
    }
    __syncthreads();

    float rs = 0.0f;
    if (lane < 16) {
        float best[4] = {BIGF, BIGF, BIGF, BIGF};
        for (int j = 0; j < 64; ++j) {
            float x = cand[lane][j];
            if (x < best[3]) {
                best[3] = x;
                if (best[3] < best[2]) { float t = best[2]; best[2] = best[3]; best[3] = t; }
                if (best[2] < best[1]) { float t = best[1]; best[1] = best[2]; best[2] = t; }
                if (best[1] < best[0]) { float t = best[0]; best[0] = best[1]; best[1] = t; }
            }
        }
#pragma unroll
        for (int j = 0; j < 4; ++j) {
            float d2   = fmaxf(best[j], EPS_C);
            float dist = sqrtf(d2);
            float w    = expf(-d2 / (H_C * H_C));
            rs += (RADIUS_C - dist) * w;
        }
    }
    rs += __shfl_xor(rs, 1, 32);
    rs += __shfl_xor(rs, 2, 32);
    rs += __shfl_xor(rs, 4, 32);
    rs += __shfl_xor(rs, 8, 32);
    rs += __shfl_xor(rs, 16, 32);
    if (lane == 0) partials[blockIdx.x] = rs;
}

// ---------------------------------------------------------------------------
// Deterministic fixed-order final reduction (double accumulators).
// ---------------------------------------------------------------------------
__global__ void __launch_bounds__(32)
finalize_kernel(const float* __restrict__ w0, const float* __restrict__ w1,
                const float* __restrict__ w2, int nblk, int BN,
                float* __restrict__ out) {
    if (threadIdx.x == 0 && blockIdx.x == 0) {
        double s0 = 0.0, s1 = 0.0, s2 = 0.0;
        for (int i = 0; i < nblk; ++i) s0 += (double)w0[i];
        for (int i = 0; i < nblk; ++i) s1 += (double)w1[i];
        for (int i = 0; i < nblk; ++i) s2 += (double)w2[i];
        double cd  = s0 / (double)BN + s1 / (double)BN;
        double rep = s2 / (double)(4 * BN);
        out[0] = (float)(cd * 100.0);
        out[1] = (float)((double)ALPHA_C * rep);
    }
}

extern "C" void kernel_launch(void* const* d_in, const int* in_sizes, int n_in,
                              void* d_out, int out_size, void* d_ws, size_t ws_size,
                              hipStream_t stream) {
    const float* pred = (const float*)d_in[0];
    const float* gt   = (const float*)d_in[1];
    // d_in[2] (pcd_radius) is unused by the reference outputs.

    const int B = in_sizes[2];                 // pcd_radius has B elements
    const int N = in_sizes[0] / (B * 3);       // 4096
    const int ntiles = N / 16;
    const int nblk = B * ntiles;               // 1024

    float* w0 = (float*)d_ws;                  // chamfer dir 1 partials
    float* w1 = w0 + nblk;                     // chamfer dir 2 partials
    float* w2 = w1 + nblk;                     // repulsion partials

    chamfer_rowmin_kernel<<<nblk, 32, 0, stream>>>(pred, gt, w0, N);
    chamfer_rowmin_kernel<<<nblk, 32, 0, stream>>>(gt, pred, w1, N);
    repulsion_kernel<<<nblk, 32, 0, stream>>>(pred, w2, N);
    finalize_kernel<<<1, 32, 0, stream>>>(w0, w1, w2, nblk, B * N, (float*)d_out);
}